// Upsample_40621800685869
// MI455X (gfx1250) — compile-verified
//
#include <hip/hip_runtime.h>

// 2x upsample (upfirdn2d, up=2, pad=(2,1), separable [1,3,3,1] kernel) for
// x: (16,256,64,64) f32 -> out: (16,256,128,128) f32.
// Memory-bound (~335 MB @ 23.3 TB/s ~= 14 us floor). Strategy: stage each
// 64x64 input plane into LDS with CDNA5 async global->LDS B128 loads
// (ASYNCcnt), then each thread emits coalesced float4 output rows via 4-tap
// FMAs reading conflict-free from LDS. Non-temporal stores (write-once).

typedef float v4f __attribute__((ext_vector_type(4)));
typedef int   v4i __attribute__((ext_vector_type(4)));

#define IN_H   64
#define IN_W   64
#define OUT_H  128
#define OUT_W  128
#define LDS_W  72   // row stride: keeps async B128 dests 16B-aligned, banks clean
#define LDS_H  66   // rows -1..64

__device__ __forceinline__ void async_load_b128(const void* g, void* l) {
#if __has_builtin(__builtin_amdgcn_global_load_async_to_lds_b128)
    // signature (probe-confirmed via diagnostic): (v4i AS1*, v4i AS3*, imm offset, imm cpol)
    __builtin_amdgcn_global_load_async_to_lds_b128(
        (__attribute__((address_space(1))) v4i*)g,
        (__attribute__((address_space(3))) v4i*)l,
        /*offset=*/0, /*cpol=*/0);
#else
    unsigned loff = (unsigned)(__UINTPTR_TYPE__)(__attribute__((address_space(3))) void*)l;
    unsigned long long ga = (unsigned long long)(__UINTPTR_TYPE__)g;
    asm volatile("global_load_async_to_lds_b128 %0, %1, off"
                 :: "v"(loff), "v"(ga) : "memory");
#endif
}

__device__ __forceinline__ void wait_async0() {
#if __has_builtin(__builtin_amdgcn_s_wait_asynccnt)
    __builtin_amdgcn_s_wait_asynccnt(0);
#else
    asm volatile("s_wait_asynccnt 0" ::: "memory");
#endif
}

__global__ __launch_bounds__(256) void Upsample_40621800685869_kernel(
    const float* __restrict__ x, const float* __restrict__ kern,
    float* __restrict__ out)
{
    __shared__ float tile[LDS_H * LDS_W];

    const int tid   = threadIdx.x;
    const int plane = blockIdx.x;                       // n*C + c
    const float* __restrict__ xp = x + (size_t)plane * (IN_H * IN_W);
    float* __restrict__       op = out + (size_t)plane * (OUT_H * OUT_W);

    // ---- 1) async-stage the 64x64 plane into LDS interior (rows 1..64, cols 4..67)
    // 4096 floats = 1024 float4 chunks; 256 threads x 4.
    #pragma unroll
    for (int k = 0; k < 4; ++k) {
        const int q  = k * 256 + tid;       // float4 id in plane
        const int r  = q >> 4;              // input row (16 float4 per row)
        const int c4 = q & 15;              // float4 column
        const void* g = (const void*)(xp + q * 4);
        void*       l = (void*)&tile[(r + 1) * LDS_W + 4 + c4 * 4];  // 16B aligned
        async_load_b128(g, l);
    }

    // ---- 2) zero the halo (row -1, row 64, col -1, col 64): 260 cells,
    // disjoint from the async destinations so no ordering hazard.
    for (int i = tid; i < 260; i += 256) {
        int idx;
        if (i < 66)        idx = 0 * LDS_W + 3 + i;             // top row, cols 3..68
        else if (i < 132)  idx = 65 * LDS_W + 3 + (i - 66);     // bottom row
        else if (i < 196)  idx = (i - 131) * LDS_W + 3;         // left col, rows 1..64
        else               idx = (i - 195) * LDS_W + 68;        // right col, rows 1..64
        tile[idx] = 0.0f;
    }

    // ---- 3) flipped kernel taps (uniform -> SGPRs). wf[a][b] = kern[3-a][3-b].
    float wf[4][4];
    #pragma unroll
    for (int a = 0; a < 4; ++a)
        #pragma unroll
        for (int b = 0; b < 4; ++b)
            wf[a][b] = kern[(3 - a) * 4 + (3 - b)];

    wait_async0();      // this wave's async LDS writes have landed
    __syncthreads();    // all waves' writes visible

    // ---- 4) compute: 128x128 outputs as 4096 float4 rows-chunks, 16/thread.
    // Output (oy, 4j..4j+3): y = oy>>1, ry = oy&1; input rows (y-1+ry, y+ry),
    // input cols 2j-1..2j+2. LDS row of first contributing row = y+ry.
    #pragma unroll 4
    for (int k = 0; k < 16; ++k) {
        const int o  = k * 256 + tid;
        const int oy = o >> 5;              // 32 float4 per output row
        const int j  = o & 31;
        const int y  = oy >> 1;
        const int ry = oy & 1;

        const float* rA = &tile[(y + ry) * LDS_W + 3 + 2 * j];
        const float* rB = rA + LDS_W;
        const float a0 = rA[0], a1 = rA[1], a2 = rA[2], a3 = rA[3];
        const float b0 = rB[0], b1 = rB[1], b2 = rB[2], b3 = rB[3];

        // row-phase weight selection (uniform cndmasks)
        const float wA0 = ry ? wf[1][0] : wf[0][0];
        const float wA1 = ry ? wf[1][1] : wf[0][1];
        const float wA2 = ry ? wf[1][2] : wf[0][2];
        const float wA3 = ry ? wf[1][3] : wf[0][3];
        const float wB0 = ry ? wf[3][0] : wf[2][0];
        const float wB1 = ry ? wf[3][1] : wf[2][1];
        const float wB2 = ry ? wf[3][2] : wf[2][2];
        const float wB3 = ry ? wf[3][3] : wf[2][3];

        v4f v;
        v.x = wA0 * a0 + wA2 * a1 + wB0 * b0 + wB2 * b1;   // ox = 4j   (rx=0)
        v.y = wA1 * a1 + wA3 * a2 + wB1 * b1 + wB3 * b2;   // ox = 4j+1 (rx=1)
        v.z = wA0 * a1 + wA2 * a2 + wB0 * b1 + wB2 * b2;   // ox = 4j+2 (rx=0)
        v.w = wA1 * a2 + wA3 * a3 + wB1 * b2 + wB3 * b3;   // ox = 4j+3 (rx=1)

        __builtin_nontemporal_store(v, (v4f*)(op + oy * OUT_W + 4 * j));
    }
}

extern "C" void kernel_launch(void* const* d_in, const int* in_sizes, int n_in,
                              void* d_out, int out_size, void* d_ws, size_t ws_size,
                              hipStream_t stream) {
    const float* x    = (const float*)d_in[0];   // (16,256,64,64) f32
    const float* kern = (const float*)d_in[1];   // (4,4) f32
    float* out        = (float*)d_out;           // (16,256,128,128) f32

    const int planes = in_sizes[0] / (IN_H * IN_W);   // 16*256 = 4096
    Upsample_40621800685869_kernel<<<dim3(planes), dim3(256), 0, stream>>>(x, kern, out);
}